// GNN_41214506172827
// MI455X (gfx1250) — compile-verified
//
#include <hip/hip_runtime.h>

// ---------------- problem constants (match reference) ----------------
#define N_NODES   50000
#define N_EDGES   1600000
#define IN_FEATS  128
#define HIDDEN    128
#define N_CLASSES 16
#define N_GRAPHS  512

typedef float v2f __attribute__((ext_vector_type(2)));
typedef float v8f __attribute__((ext_vector_type(8)));
typedef unsigned int v4u __attribute__((ext_vector_type(4)));
typedef unsigned int v8u __attribute__((ext_vector_type(8)));

// ---------------- TDM: 1D tensor load (n dwords) into LDS ----------------
// Builds a Tensor DMA Descriptor (D#) per CDNA5 ISA ch.8 and issues
// tensor_load_to_lds with groups 0+1 (<=2D tensor, groups 2/3 omitted).
// Call from ONE wave only; follow with s_wait_tensorcnt + block barrier.
__device__ __forceinline__ void tdm_load_1d_to_lds(const void* gptr, unsigned lds_byte_off,
                                                   unsigned n_dwords) {
    unsigned long long ga = (unsigned long long)(uintptr_t)gptr;
    v4u g0;
    g0[0] = 1u;                                              // count=1 (valid), user D#
    g0[1] = lds_byte_off;                                    // lds_addr (bytes)
    g0[2] = (unsigned)(ga & 0xffffffffu);                    // global_addr[31:0]
    g0[3] = (unsigned)((ga >> 32) & 0x01ffffffu)             // global_addr[56:32]
            | 0x80000000u;                                   // type=2 ("image")
    v8u g1;
    g1[0] = 2u << 16;                                        // wg_mask=0, data_size=2 (4B)
    g1[1] = (n_dwords & 0xffffu) << 16;                      // tensor_dim0[15:0]
    g1[2] = ((n_dwords >> 16) & 0xffffu) | (1u << 16);       // tensor_dim0[31:16], tensor_dim1=1
    g1[3] = (n_dwords & 0xffffu) << 16;                      // tile_dim0 (bits 127:112)
    g1[4] = 0;                                               // tile_dim1=0, tile_dim2=0 (1D)
    g1[5] = n_dwords;                                        // tensor_dim0_stride[31:0]
    g1[6] = 0;                                               // stride hi / dim1_stride lo
    g1[7] = 0;                                               // dim1_stride hi
    asm volatile("tensor_load_to_lds %0, %1" :: "s"(g0), "s"(g1) : "memory");
}

// ---------------- zero fill (grid-stride) ----------------
__global__ void zero_f32(float* __restrict__ p, long n) {
    long i = (long)blockIdx.x * blockDim.x + threadIdx.x;
    long stride = (long)gridDim.x * blockDim.x;
    for (; i < n; i += stride) p[i] = 0.0f;
}

// ---------------- degree histogram over edges ----------------
__global__ void degree_kernel(const int* __restrict__ src, const int* __restrict__ dst,
                              float* __restrict__ deg_out, float* __restrict__ deg_in,
                              int n_edges) {
    int e = blockIdx.x * blockDim.x + threadIdx.x;
    if (e < n_edges) {
        atomicAdd(&deg_out[src[e]], 1.0f);
        atomicAdd(&deg_in[dst[e]], 1.0f);
    }
}

// ---------------- norm = deg>0 ? rsqrt(max(deg,1)) : 0 ----------------
__global__ void norm_kernel(const float* __restrict__ deg, float* __restrict__ norm, int n) {
    int i = blockIdx.x * blockDim.x + threadIdx.x;
    if (i < n) {
        float d = deg[i];
        norm[i] = (d > 0.0f) ? rsqrtf(fmaxf(d, 1.0f)) : 0.0f;
    }
}

// ---------------- SpMM: agg[dst] += x[src] * norm_out[src] ----------------
// one wave (32 lanes) per edge; b128 gather + 4x global_atomic_add_f32
// (accumulator is 25.6MB -> resident in the 192MB L2, atomics resolve there).
__global__ void spmm_kernel(const float* __restrict__ x, const float* __restrict__ norm_out,
                            const int* __restrict__ src, const int* __restrict__ dst,
                            float* __restrict__ agg, int n_edges) {
    int wave = threadIdx.x >> 5;
    int lane = threadIdx.x & 31;
    int e = blockIdx.x * (blockDim.x >> 5) + wave;
    if (e >= n_edges) return;
    int s = src[e];
    int d = dst[e];
    float w = norm_out[s];
    const float4* xr = (const float4*)(x + (long)s * HIDDEN);
    float4 v = xr[lane];
    float* ad = agg + (long)d * HIDDEN + lane * 4;
    atomicAdd(ad + 0, v.x * w);
    atomicAdd(ad + 1, v.y * w);
    atomicAdd(ad + 2, v.z * w);
    atomicAdd(ad + 3, v.w * w);
}

// ---------------- GEMM+bias+ReLU with V_WMMA_F32_16X16X4_F32 ----------------
// out[M x 128] = relu( (A * norm_in[row]) @ W[128x128] + bias )
// W (64KB) is staged into LDS once per block via the Tensor Data Mover;
// one wave per 16-row strip, 8 N-tiles x 32 WMMAs, B-fragments from LDS.
__global__ void gemm_relu_kernel(const float* __restrict__ A, const float* __restrict__ norm_in,
                                 const float* __restrict__ W, const float* __restrict__ bias,
                                 float* __restrict__ out, int nrows) {
    extern __shared__ float sW[];                    // 128*128 f32 = 64KB dynamic LDS

    if (threadIdx.x < 32) {                          // wave 0 issues the TDM copy
        unsigned lds_off = (unsigned)(unsigned long long)(uintptr_t)sW; // low 32 = LDS byte off
        tdm_load_1d_to_lds(W, lds_off, HIDDEN * HIDDEN);
        __builtin_amdgcn_s_wait_tensorcnt(0);        // per-wave TENSORcnt
    }
    __syncthreads();                                 // publish LDS tile to all 8 waves

    int lane = threadIdx.x & 31;
    int wave = threadIdx.x >> 5;
    int strip = blockIdx.x * (blockDim.x >> 5) + wave;
    if (strip * 16 >= nrows) return;                 // wave-uniform: EXEC all-1s for WMMA
    int rowBase = strip * 16;
    int half = lane >> 4;                            // K sub-pair select
    int idx  = lane & 15;

    int arow = rowBase + idx;                        // A: lane holds M = idx
    float scale = norm_in[arow];
    const float* ap = A + (long)arow * HIDDEN;

    v2f aFrag[32];
#pragma unroll
    for (int s = 0; s < 32; ++s) {
        int k = 4 * s + 2 * half;
        aFrag[s].x = ap[k]     * scale;
        aFrag[s].y = ap[k + 1] * scale;
    }

    for (int nt = 0; nt < 8; ++nt) {
        int n = nt * 16 + idx;                       // B/C: lane holds N = idx
        v8f c = {};
#pragma unroll
        for (int s = 0; s < 32; ++s) {
            int k = 4 * s + 2 * half;
            v2f b;
            b.x = sW[k * HIDDEN + n];                // ds_load from staged W
            b.y = sW[(k + 1) * HIDDEN + n];
            c = __builtin_amdgcn_wmma_f32_16x16x4_f32(
                    false, aFrag[s], false, b, (short)0, c, false, false);
        }
        float bn = bias[n];
#pragma unroll
        for (int r = 0; r < 8; ++r) {
            int m = rowBase + r + 8 * half;          // C layout: VGPR r -> M = r + 8*half
            float v = c[r] + bn;
            out[(long)m * HIDDEN + n] = v > 0.0f ? v : 0.0f;
        }
    }
}

// ---------------- per-graph mean pooling (sum + count via atomics) -------
__global__ void pool_kernel(const float* __restrict__ x, const int* __restrict__ gid,
                            float* __restrict__ pooled, float* __restrict__ counts) {
    int node = blockIdx.x;
    int f = threadIdx.x;                             // 128 threads
    int g = gid[node];
    atomicAdd(&pooled[(long)g * HIDDEN + f], x[(long)node * HIDDEN + f]);
    if (f == 0) atomicAdd(&counts[g], 1.0f);
}

// ---------------- classifier: (pooled/count) @ Wf + bf, N=16, WMMA f32 ----
__global__ void final_gemm_kernel(const float* __restrict__ pooled, const float* __restrict__ counts,
                                  const float* __restrict__ Wf, const float* __restrict__ bf,
                                  float* __restrict__ out) {
    int lane = threadIdx.x & 31;
    int strip = blockIdx.x;                          // 32 strips of 16 graphs
    int rowBase = strip * 16;
    int half = lane >> 4;
    int idx  = lane & 15;

    int g = rowBase + idx;
    float scale = 1.0f / fmaxf(counts[g], 1.0f);
    const float* ap = pooled + (long)g * HIDDEN;

    v8f c = {};
#pragma unroll
    for (int s = 0; s < 32; ++s) {
        int k = 4 * s + 2 * half;
        v2f a, b;
        a.x = ap[k]     * scale;
        a.y = ap[k + 1] * scale;
        b.x = Wf[(long)k * N_CLASSES + idx];
        b.y = Wf[(long)(k + 1) * N_CLASSES + idx];
        c = __builtin_amdgcn_wmma_f32_16x16x4_f32(
                false, a, false, b, (short)0, c, false, false);
    }
    float bn = bf[idx];
#pragma unroll
    for (int r = 0; r < 8; ++r) {
        int m = rowBase + r + 8 * half;
        out[(long)m * N_CLASSES + idx] = c[r] + bn;
    }
}

// ---------------- host orchestration ----------------
extern "C" void kernel_launch(void* const* d_in, const int* in_sizes, int n_in,
                              void* d_out, int out_size, void* d_ws, size_t ws_size,
                              hipStream_t stream) {
    const float* features = (const float*)d_in[0];
    const float* W1 = (const float*)d_in[1];
    const float* b1 = (const float*)d_in[2];
    const float* W2 = (const float*)d_in[3];
    const float* b2 = (const float*)d_in[4];
    const float* Wf = (const float*)d_in[5];
    const float* bfv = (const float*)d_in[6];
    const int* src = (const int*)d_in[7];
    const int* dst = (const int*)d_in[8];
    const int* gids = (const int*)d_in[9];
    float* out = (float*)d_out;

    // workspace layout (floats)
    float* ws = (float*)d_ws;
    float* deg_out  = ws;                                   // 50000
    float* deg_in   = deg_out + N_NODES;                    // 50000
    float* pooled   = deg_in + N_NODES;                     // 512*128
    float* counts   = pooled + (long)N_GRAPHS * HIDDEN;     // 512
    float* norm_out = counts + N_GRAPHS;                    // 50000
    float* norm_in  = norm_out + N_NODES;                   // 50000
    float* bufA     = norm_in + N_NODES;                    // 50000*128 (agg)
    float* bufB     = bufA + (long)N_NODES * HIDDEN;        // 50000*128 (activations)

    const long actElems = (long)N_NODES * HIDDEN;
    const long zeroHead = 2L * N_NODES + (long)N_GRAPHS * HIDDEN + N_GRAPHS;
    const unsigned ldsBytes = HIDDEN * HIDDEN * sizeof(float);  // 64KB W stage

    // 0) zero accumulators (deg/pooled/counts block + agg buffer A)
    zero_f32<<<1024, 256, 0, stream>>>(deg_out, zeroHead);
    zero_f32<<<2048, 256, 0, stream>>>(bufA, actElems);

    // 1) degrees + norms
    degree_kernel<<<(N_EDGES + 255) / 256, 256, 0, stream>>>(src, dst, deg_out, deg_in, N_EDGES);
    norm_kernel<<<(N_NODES + 255) / 256, 256, 0, stream>>>(deg_out, norm_out, N_NODES);
    norm_kernel<<<(N_NODES + 255) / 256, 256, 0, stream>>>(deg_in, norm_in, N_NODES);

    // 2) layer 1: SpMM then fused scale+GEMM+bias+ReLU (TDM + WMMA f32)
    spmm_kernel<<<(N_EDGES + 7) / 8, 256, 0, stream>>>(features, norm_out, src, dst, bufA, N_EDGES);
    gemm_relu_kernel<<<(N_NODES / 16 + 7) / 8, 256, ldsBytes, stream>>>(bufA, norm_in, W1, b1, bufB, N_NODES);

    // 3) layer 2
    zero_f32<<<2048, 256, 0, stream>>>(bufA, actElems);
    spmm_kernel<<<(N_EDGES + 7) / 8, 256, 0, stream>>>(bufB, norm_out, src, dst, bufA, N_EDGES);
    gemm_relu_kernel<<<(N_NODES / 16 + 7) / 8, 256, ldsBytes, stream>>>(bufA, norm_in, W2, b2, bufB, N_NODES);

    // 4) mean pool + classifier GEMM (WMMA f32, N=16)
    pool_kernel<<<N_NODES, HIDDEN, 0, stream>>>(bufB, gids, pooled, counts);
    final_gemm_kernel<<<N_GRAPHS / 16, 32, 0, stream>>>(pooled, counts, Wf, bfv, out);
}